// testModel_22960895164550
// MI455X (gfx1250) — compile-verified
//
#include <hip/hip_runtime.h>

typedef __attribute__((ext_vector_type(2))) float v2f;
typedef __attribute__((ext_vector_type(8))) float v8f;
typedef __attribute__((ext_vector_type(4))) unsigned int u32x4;
typedef __attribute__((ext_vector_type(4))) int i32x4;
typedef __attribute__((ext_vector_type(8))) int i32x8;

#define BN_EPS 1e-5f

#if defined(__HIP_DEVICE_COMPILE__) && defined(__has_builtin)
#if __has_builtin(__builtin_amdgcn_tensor_load_to_lds) && \
    __has_builtin(__builtin_amdgcn_s_wait_tensorcnt)
#define HAVE_TDM 1
#endif
#endif

// ---------------------------------------------------------------------------
// Elementwise / edge kernels
// ---------------------------------------------------------------------------
__global__ void k_zero(float* __restrict__ p, int n) {
  int i = blockIdx.x * blockDim.x + threadIdx.x;
  if (i < n) p[i] = 0.0f;
}

__global__ void k_count(const int* __restrict__ dst, int E, float* __restrict__ cnt) {
  int e = blockIdx.x * blockDim.x + threadIdx.x;
  if (e < E) atomicAdd(&cnt[dst[e]], 1.0f);
}

__global__ void k_recip(const float* __restrict__ cnt, float* __restrict__ rs, int n) {
  int i = blockIdx.x * blockDim.x + threadIdx.x;
  if (i < n) rs[i] = 1.0f / fmaxf(cnt[i], 1.0f);
}

// scatter-add 16-dim per-node features along edges: agg[dst] += feat[src]
__global__ void k_scatter16(const int* __restrict__ src, const int* __restrict__ dst,
                            int total, const float* __restrict__ feat,
                            float* __restrict__ agg) {
  int i = blockIdx.x * blockDim.x + threadIdx.x;
  if (i < total) {
    int e = i >> 4, c = i & 15;
    atomicAdd(&agg[dst[e] * 16 + c], feat[src[e] * 16 + c]);
  }
}

// y1 = relu(agg*rs + q + bl[c])   (sage1 epilogue), 16 channels
__global__ void k_finish1(const float* __restrict__ agg, const float* __restrict__ rs,
                          const float* __restrict__ q, const float* __restrict__ bl,
                          float* __restrict__ y, int total) {
  int i = blockIdx.x * blockDim.x + threadIdx.x;
  if (i < total) {
    int n = i >> 4, c = i & 15;
    y[i] = fmaxf(agg[i] * rs[n] + q[i] + bl[c], 0.0f);
  }
}

// ---------------------------------------------------------------------------
// TDM weight staging: DMA W[i*wstride], i in [0, nelem) into LDS as packed f32.
// D# per CDNA5 ISA ch.8: group0 = {flags, lds_addr, global_addr, type=2},
// group1 = dims/strides. wstride==1 -> one contiguous row; wstride==2 ->
// nelem rows of one element with tensor_dim0_stride=2 (tap-0 gather).
// ---------------------------------------------------------------------------
#ifdef HAVE_TDM
__device__ __forceinline__ void tdm_load_weights(const float* __restrict__ Wg,
                                                 float* __restrict__ sW,
                                                 int nelem, int wstride) {
  unsigned long long ga = (unsigned long long)(size_t)Wg;
  unsigned lds = (unsigned)(size_t)sW;  // low 32 bits of shared aperture = LDS byte addr
  unsigned td0, td1, t0, t1;
  unsigned long long s0;
  if (wstride == 1) { td0 = (unsigned)nelem; td1 = 1u; t0 = (unsigned)nelem; t1 = 1u; s0 = (unsigned long long)nelem; }
  else              { td0 = 1u; td1 = (unsigned)nelem; t0 = 1u; t1 = (unsigned)nelem; s0 = (unsigned long long)wstride; }

  u32x4 g0;
  g0[0] = 1u;                                               // count=1, user descriptor
  g0[1] = lds;                                              // lds_addr
  g0[2] = (unsigned)(ga & 0xFFFFFFFFull);                   // global_addr[31:0]
  g0[3] = (unsigned)((ga >> 32) & 0x1FFFFFFull) | (2u << 30);  // addr[56:32] | type=2

  i32x8 g1;
  g1[0] = (int)(2u << 16);                                  // data_size=4B, no multicast
  g1[1] = (int)((td0 & 0xFFFFu) << 16);                     // tensor_dim0[15:0]
  g1[2] = (int)(((td0 >> 16) & 0xFFFFu) | ((td1 & 0xFFFFu) << 16));
  g1[3] = (int)(((td1 >> 16) & 0xFFFFu) | ((t0 & 0xFFFFu) << 16));   // tile_dim0
  g1[4] = (int)(t1 & 0xFFFFu);                              // tile_dim1, tile_dim2=0
  g1[5] = (int)(unsigned)(s0 & 0xFFFFFFFFull);              // tensor_dim0_stride[31:0]
  g1[6] = (int)(unsigned)((s0 >> 32) & 0xFFFFull);          // stride[47:32], dim1_stride=0
  g1[7] = 0;

  i32x4 z4 = {0, 0, 0, 0};
#if __clang_major__ >= 23
  i32x8 z8 = {0, 0, 0, 0, 0, 0, 0, 0};
  __builtin_amdgcn_tensor_load_to_lds(g0, g1, z4, z4, z8, 0);
#else
  __builtin_amdgcn_tensor_load_to_lds(g0, g1, z4, z4, 0);
#endif
}
#endif

// ---------------------------------------------------------------------------
// Generic fused linear stage with fp32 WMMA (V_WMMA_F32_16X16X4_F32):
//   Y[N, COUT] = post( pre(X[N, CIN]) @ W^T + bias ) + pad_residual(R)
// Each wave handles TWO 16-row tiles (32 rows): every B fragment fetched from
// LDS feeds two WMMAs. A (global), B and pre-BN scale/shift (LDS) are
// double-buffered one k-step ahead. PRERELU is compile-time: no control flow
// inside the WMMA loop.
// ---------------------------------------------------------------------------
template <int CIN, int COUT, bool PRERELU>
__global__ __launch_bounds__(256) void k_lin(
    const float* __restrict__ X,
    const float* __restrict__ rowscale,                       // nullable per-row multiplier
    const float* preG, const float* preB,
    const float* preM, const float* preV,                     // pre-BN (nullable G)
    const float* __restrict__ Wg, int wstride,
    const float* __restrict__ bias,                           // nullable
    const float* postG, const float* postB,
    const float* postM, const float* postV, bool postrelu,    // post-BN (nullable G)
    const float* __restrict__ R, int ch1, int cres,           // residual (nullable)
    float* __restrict__ Y, int nrows) {
  extern __shared__ float smem[];
  float* sW  = smem;                 // [COUT][CIN] packed weights
  float* sPS = smem + CIN * COUT;    // pre scale  [CIN]
  float* sPT = sPS + CIN;            // pre shift  [CIN]

#ifdef HAVE_TDM
  if ((threadIdx.x >> 5) == 0) tdm_load_weights(Wg, sW, CIN * COUT, wstride);
#else
  for (int i = threadIdx.x; i < CIN * COUT; i += blockDim.x)
    sW[i] = Wg[i * wstride];
#endif
  for (int i = threadIdx.x; i < CIN; i += blockDim.x) {
    float s = 1.0f, t = 0.0f;
    if (preG) {
      s = preG[i] * __frsqrt_rn(preV[i] + BN_EPS);
      t = preB[i] - preM[i] * s;
    }
    sPS[i] = s; sPT[i] = t;
  }
#ifdef HAVE_TDM
  if ((threadIdx.x >> 5) == 0) __builtin_amdgcn_s_wait_tensorcnt(0);
#endif
  __syncthreads();

  const int lane = threadIdx.x & 31;
  const int wave = threadIdx.x >> 5;
  const int pair = blockIdx.x * (blockDim.x >> 5) + wave;
  const int row0 = pair * 32;                 // two 16-row tiles per wave
  if (row0 >= nrows) return;  // wave-uniform: EXEC stays all-ones for WMMA
  const bool has1 = (row0 + 16) < nrows;

  const int kk = lane >> 4;   // K half (A/B fragment layout)
  const int ln = lane & 15;   // M (A) / N (B,C,D)
  const int rowa0 = row0 + ln;
  const int rowa1 = has1 ? (row0 + 16 + ln) : rowa0;
  const float rs0 = rowscale ? rowscale[rowa0] : 1.0f;
  const float rs1 = rowscale ? rowscale[rowa1] : 1.0f;

  constexpr int NCT = COUT / 16;
  v8f acc0[NCT], acc1[NCT];
#pragma unroll
  for (int ct = 0; ct < NCT; ++ct)
#pragma unroll
    for (int j = 0; j < 8; ++j) { acc0[ct][j] = 0.0f; acc1[ct][j] = 0.0f; }

  const float* xr0 = X + (size_t)rowa0 * CIN;
  const float* xr1 = X + (size_t)rowa1 * CIN;

  // ---- prologue: preload k-step 0 operands ----
  v2f xcur0 = *(const v2f*)(xr0 + 2 * kk);
  v2f xcur1 = *(const v2f*)(xr1 + 2 * kk);
  v2f pscur, ptcur;
  pscur.x = sPS[2 * kk];     pscur.y = sPS[2 * kk + 1];
  ptcur.x = sPT[2 * kk];     ptcur.y = sPT[2 * kk + 1];
  v2f bcur[NCT];
#pragma unroll
  for (int ct = 0; ct < NCT; ++ct) {
    const float* wp = sW + (ct * 16 + ln) * CIN + 2 * kk;
    bcur[ct].x = wp[0]; bcur[ct].y = wp[1];
  }

#pragma unroll
  for (int k0 = 0; k0 < CIN; k0 += 4) {
    const int kn = (k0 + 4 < CIN) ? (k0 + 4) : 0;  // wrap on last (result unused)
    // ---- prefetch next k-step ----
    v2f xnext0 = *(const v2f*)(xr0 + kn + 2 * kk);
    v2f xnext1 = *(const v2f*)(xr1 + kn + 2 * kk);
    v2f psn, ptn;
    psn.x = sPS[kn + 2 * kk]; psn.y = sPS[kn + 2 * kk + 1];
    ptn.x = sPT[kn + 2 * kk]; ptn.y = sPT[kn + 2 * kk + 1];
    v2f bnext[NCT];
#pragma unroll
    for (int ct = 0; ct < NCT; ++ct) {
      const float* wp = sW + (ct * 16 + ln) * CIN + kn + 2 * kk;
      bnext[ct].x = wp[0]; bnext[ct].y = wp[1];
    }
    // ---- compute with current operands (no branches) ----
    float a0x = pscur.x * xcur0.x + ptcur.x;
    float a0y = pscur.y * xcur0.y + ptcur.y;
    float a1x = pscur.x * xcur1.x + ptcur.x;
    float a1y = pscur.y * xcur1.y + ptcur.y;
    if (PRERELU) {
      a0x = fmaxf(a0x, 0.0f); a0y = fmaxf(a0y, 0.0f);
      a1x = fmaxf(a1x, 0.0f); a1y = fmaxf(a1y, 0.0f);
    }
    v2f a0; a0.x = a0x * rs0; a0.y = a0y * rs0;
    v2f a1; a1.x = a1x * rs1; a1.y = a1y * rs1;
#pragma unroll
    for (int ct = 0; ct < NCT; ++ct) {   // B reused by two WMMAs back-to-back
      acc0[ct] = __builtin_amdgcn_wmma_f32_16x16x4_f32(
          false, a0, false, bcur[ct], (short)0, acc0[ct], false, false);
      acc1[ct] = __builtin_amdgcn_wmma_f32_16x16x4_f32(
          false, a1, false, bcur[ct], (short)0, acc1[ct], false, false);
    }
    // ---- rotate buffers ----
    xcur0 = xnext0; xcur1 = xnext1; pscur = psn; ptcur = ptn;
#pragma unroll
    for (int ct = 0; ct < NCT; ++ct) bcur[ct] = bnext[ct];
  }

  // Epilogue: bias, post-BN(+ReLU), zero-padded residual, store (both tiles).
#pragma unroll
  for (int ct = 0; ct < NCT; ++ct) {
    const int col = ct * 16 + ln;
    const float bv = bias ? bias[col] : 0.0f;
    float ps = 1.0f, pt = 0.0f;
    if (postG) {
      ps = postG[col] * __frsqrt_rn(postV[col] + BN_EPS);
      pt = postB[col] - postM[col] * ps;
    }
    const int rc = col - ch1;
    const bool dores = (R != nullptr) && (rc >= 0) && (rc < cres);
#pragma unroll
    for (int j = 0; j < 8; ++j) {
      const int r = row0 + j + 8 * kk;  // C/D layout: M = vgpr + 8*(lane>=16)
      float v = acc0[ct][j] + bv;
      v = ps * v + pt;
      if (postrelu) v = fmaxf(v, 0.0f);
      if (dores) v += R[(size_t)r * cres + rc];
      Y[(size_t)r * COUT + col] = v;
    }
    if (has1) {
#pragma unroll
      for (int j = 0; j < 8; ++j) {
        const int r = row0 + 16 + j + 8 * kk;
        float v = acc1[ct][j] + bv;
        v = ps * v + pt;
        if (postrelu) v = fmaxf(v, 0.0f);
        if (dores) v += R[(size_t)r * cres + rc];
        Y[(size_t)r * COUT + col] = v;
      }
    }
  }
}

// ---------------------------------------------------------------------------
// Host side
// ---------------------------------------------------------------------------
struct LinArgs {
  const float* X = nullptr;
  const float* rowscale = nullptr;
  const float *preG = nullptr, *preB = nullptr, *preM = nullptr, *preV = nullptr;
  bool prerelu = false;
  const float* W = nullptr; int wstride = 1;
  const float* bias = nullptr;
  const float *postG = nullptr, *postB = nullptr, *postM = nullptr, *postV = nullptr;
  bool postrelu = false;
  const float* R = nullptr; int ch1 = 0; int cres = 0;
  float* Y = nullptr; int n = 0;
};

template <int CIN, int COUT, bool PRERELU>
static void launch_lin_t(const LinArgs& a, hipStream_t s) {
  const int pairs = (a.n + 31) / 32;     // 32 rows per wave
  const int blocks = (pairs + 7) / 8;    // 8 waves (256 threads) per workgroup
  const size_t sh = (size_t)(CIN * COUT + 2 * CIN) * sizeof(float);
  k_lin<CIN, COUT, PRERELU><<<blocks, 256, sh, s>>>(
      a.X, a.rowscale, a.preG, a.preB, a.preM, a.preV,
      a.W, a.wstride, a.bias, a.postG, a.postB, a.postM, a.postV, a.postrelu,
      a.R, a.ch1, a.cres, a.Y, a.n);
}

template <int CIN, int COUT>
static void launch_lin(const LinArgs& a, hipStream_t s) {
  if (a.prerelu) launch_lin_t<CIN, COUT, true>(a, s);
  else           launch_lin_t<CIN, COUT, false>(a, s);
}

static void launch_lin_dyn(int cin, int cout, const LinArgs& a, hipStream_t s) {
  if (cin == 64 && cout == 64)        launch_lin<64, 64>(a, s);
  else if (cin == 64 && cout == 128)  launch_lin<64, 128>(a, s);
  else if (cin == 128 && cout == 128) launch_lin<128, 128>(a, s);
}

extern "C" void kernel_launch(void* const* d_in, const int* in_sizes, int n_in,
                              void* d_out, int out_size, void* d_ws, size_t ws_size,
                              hipStream_t stream) {
  (void)n_in; (void)out_size; (void)ws_size;
  const float* x  = (const float*)d_in[0];          // [N,128]
  const int*   ei = (const int*)d_in[1];            // [2,E]
  const int N = in_sizes[0] / 128;
  const int E = in_sizes[1] / 2;
  const int* src = ei;
  const int* dst = ei + E;

  auto F = [&](int i) { return (const float*)d_in[i]; };
  // params, dict insertion order:
  const float* s1Wl = F(2);  const float* s1bl = F(3);  const float* s1Wr = F(4);
  const float* s2Wl = F(5);  const float* s2bl = F(6);  const float* s2Wr = F(7);
  const float* fw = F(8);    const float* fb = F(9);
  const float *fg = F(10), *fbe = F(11), *fm = F(12), *fv = F(13);
  // blocks at 14 + 12*i: w1,b1,bn1{g,b,m,v},w2,b2,bn2{g,b,m,v}
  const float *Fg = F(110), *Fbe = F(111), *Fm = F(112), *Fv = F(113);  // final_bn
  const float* dw = F(114);  const float* db = F(115);

  // workspace layout (floats): cnt | agg1 | agg2 | rs | p16 | q16 | y1 | bufA | bufT
  float* ws = (float*)d_ws;
  size_t off = 0;
  float* cnt  = ws + off; off += (size_t)N;
  float* agg1 = ws + off; off += (size_t)N * 16;
  float* agg2 = ws + off; off += (size_t)N * 16;
  float* rs   = ws + off; off += (size_t)N;
  float* p16  = ws + off; off += (size_t)N * 16;
  float* q16  = ws + off; off += (size_t)N * 16;
  float* y1   = ws + off; off += (size_t)N * 16;
  float* bufA = ws + off; off += (size_t)N * 128;
  float* bufT = ws + off; off += (size_t)N * 128;
  float* tmp32 = p16;  // 32N floats spanning p16+q16 (both free when it's used)

  const int zn = N * 33;  // cnt + agg1 + agg2 (contiguous)
  k_zero<<<(zn + 255) / 256, 256, 0, stream>>>(cnt, zn);

  // sage1: project to 16 dims FIRST (segment-mean commutes with linear) -> 8x less edge traffic
  { LinArgs a; a.X = x; a.W = s1Wl; a.Y = p16; a.n = N; launch_lin<128, 16>(a, stream); }
  { LinArgs a; a.X = x; a.W = s1Wr; a.Y = q16; a.n = N; launch_lin<128, 16>(a, stream); }

  k_count<<<(E + 255) / 256, 256, 0, stream>>>(dst, E, cnt);
  k_recip<<<(N + 255) / 256, 256, 0, stream>>>(cnt, rs, N);

  const int Esc = E * 16;
  k_scatter16<<<(Esc + 255) / 256, 256, 0, stream>>>(src, dst, Esc, p16, agg1);
  k_finish1<<<(N * 16 + 255) / 256, 256, 0, stream>>>(agg1, rs, q16, s1bl, y1, N * 16);
  k_scatter16<<<(Esc + 255) / 256, 256, 0, stream>>>(src, dst, Esc, y1, agg2);

  // sage2: h32 = (agg2/cnt)@Wl2^T + bl2 + y1@Wr2^T   -> bufT [N,32]
  { LinArgs a; a.X = agg2; a.rowscale = rs; a.W = s2Wl; a.bias = s2bl;
    a.Y = tmp32; a.n = N; launch_lin<16, 32>(a, stream); }
  { LinArgs a; a.X = y1; a.W = s2Wr; a.R = tmp32; a.ch1 = 0; a.cres = 32;
    a.Y = bufT; a.n = N; launch_lin<16, 32>(a, stream); }

  // first conv (tap 0) + BN + ReLU: bufT[N,32] -> bufA[N,64]
  { LinArgs a; a.X = bufT; a.W = fw; a.wstride = 2; a.bias = fb;
    a.postG = fg; a.postB = fbe; a.postM = fm; a.postV = fv; a.postrelu = true;
    a.Y = bufA; a.n = N; launch_lin<32, 64>(a, stream); }

  // 8 residual blocks, ping-pong bufA <-> d_out, scratch bufT
  const int cins[8]  = {64, 64, 64, 64, 64, 128, 128, 128};
  const int couts[8] = {64, 64, 64, 64, 128, 128, 128, 128};
  float* H = bufA;
  float* other = (float*)d_out;
  for (int i = 0; i < 8; ++i) {
    const int cin = cins[i], cout = couts[i];
    const int base = 14 + 12 * i;
    const float* w1 = F(base + 0); const float* b1 = F(base + 1);
    const float *g1 = F(base + 2), *be1 = F(base + 3), *m1 = F(base + 4), *v1 = F(base + 5);
    const float* w2 = F(base + 6); const float* b2 = F(base + 7);
    const float *g2 = F(base + 8), *be2 = F(base + 9), *m2 = F(base + 10), *v2 = F(base + 11);

    // S_a: t = relu(bn2( pre(H) @ w1^T + b1 ))  (pre = bn1+relu except block 0)
    { LinArgs a; a.X = H;
      if (i > 0) { a.preG = g1; a.preB = be1; a.preM = m1; a.preV = v1; a.prerelu = true; }
      a.W = w1; a.wstride = 2; a.bias = b1;
      a.postG = g2; a.postB = be2; a.postM = m2; a.postV = v2; a.postrelu = true;
      a.Y = bufT; a.n = N;
      launch_lin_dyn(cin, cout, a, stream); }
    // S_b: H' = t @ w2^T + b2 + zero_pad(H)
    { LinArgs a; a.X = bufT; a.W = w2; a.wstride = 2; a.bias = b2;
      a.R = H; a.ch1 = (cout - cin) / 2; a.cres = cin;
      a.Y = other; a.n = N;
      launch_lin_dyn(cout, cout, a, stream); }
    float* t = H; H = other; other = t;
  }
  // after 8 blocks H == bufA

  // final: out = relu(bn_final(H)) @ dense.w^T + dense.b   -> d_out [N,128]
  { LinArgs a; a.X = H;
    a.preG = Fg; a.preB = Fbe; a.preM = Fm; a.preV = Fv; a.prerelu = true;
    a.W = dw; a.wstride = 1; a.bias = db;
    a.Y = (float*)d_out; a.n = N;
    launch_lin<128, 128>(a, stream); }
}